// HyenaFilter_28896539968137
// MI455X (gfx1250) — compile-verified
//
#include <hip/hip_runtime.h>
#include <math.h>

// ---------------------------------------------------------------------------
// Hyena filter + FFT convolution for MI455X (gfx1250), wave32, WMMA f32.
//
// Conv realized as four-step FFT (16384 = 128 x 128): both FFT stages are
// 128x128 complex matmuls against the shared DFT matrix F128, executed with
// V_WMMA_F32_16X16X4_F32. Whole per-channel FFT lives in LDS (262144 B
// ping-pong, within the 320KB/WGP budget). Signal staging uses the CDNA5
// async global->LDS path (GLOBAL_LOAD_ASYNC_TO_LDS_B128 + s_wait_asynccnt).
// ---------------------------------------------------------------------------

typedef float v2f __attribute__((ext_vector_type(2)));
typedef float v8f __attribute__((ext_vector_type(8)));

#define DIMC   768
#define LSEQ   8192
#define NFFT   16384
#define RAD    128
#define EMBD   33
#define BANDS  16
#define FO     64

#define TWOPI      6.28318530717958647692f
#define TWOPI_N    (TWOPI / (float)NFFT)
#define TWOPI_R    (TWOPI / (float)RAD)
// |deltas| endpoints: -ln(0.01)/1.5 and -ln(0.01)/0.3
#define MIN_AD 3.0701134573253944f
#define MAX_AD 15.350567286626972f

// ---- WMMA f32 16x16x4 wrapper ---------------------------------------------
__device__ __forceinline__ v8f wmma4(v2f a, v2f b, v8f c) {
  // (neg_a, A, neg_b, B, c_mod, C, reuse_a, reuse_b)
  return __builtin_amdgcn_wmma_f32_16x16x4_f32(false, a, false, b, (short)0, c,
                                               false, false);
}

// A fragment (16x4): lane L holds M = L%16; VGPR v holds K = v + 2*(L/16).
__device__ __forceinline__ v2f frag_A(const float* p, int r0, int c0, int ld,
                                      float sgn) {
  const int L = threadIdx.x & 31;
  const int m = L & 15;
  const int c = c0 + ((L >> 4) << 1);
  const float* q = p + (r0 + m) * ld + c;
  v2f f;
  f.x = sgn * q[0];
  f.y = sgn * q[1];
  return f;
}

// B fragment (4x16): lane L holds N = L%16; VGPR v holds K = v + 2*(L/16).
__device__ __forceinline__ v2f frag_B(const float* p, int r0, int c0, int ld,
                                      float sgn) {
  const int L = threadIdx.x & 31;
  const int n = L & 15;
  const int r = r0 + ((L >> 4) << 1);
  v2f f;
  f.x = sgn * p[r * ld + c0 + n];
  f.y = sgn * p[(r + 1) * ld + c0 + n];
  return f;
}

// ---- Async global->LDS staging (ASYNCcnt path) -----------------------------
// Copies 'nfloats' f32 (multiple of 1024 = 256 threads * 4) from uniform
// 'gsrc' into LDS at 'ldst' with GLOBAL_LOAD_ASYNC_TO_LDS_B128 (GVS mode:
// SGPR64 base + per-lane 32-bit byte offset).
__device__ __forceinline__ void async_copy_to_lds(float* ldst,
                                                  const float* gsrc,
                                                  int nfloats) {
  unsigned long long gbase = (unsigned long long)(const void*)gsrc;
  for (int idx4 = ((int)threadIdx.x) << 2; idx4 < nfloats; idx4 += 256 * 4) {
    unsigned lds_addr = (unsigned)(uintptr_t)(void*)(ldst + idx4);
    unsigned goff = (unsigned)idx4 << 2;  // byte offset
    asm volatile("global_load_async_to_lds_b128 %0, %1, %2"
                 :
                 : "v"(lds_addr), "v"(goff), "s"(gbase)
                 : "memory");
  }
}
__device__ __forceinline__ void wait_async() {
  asm volatile("s_wait_asynccnt 0" ::: "memory");
}

// ---- Stage: C = F~ * X  (left multiply), optional twiddle, write to dst ----
__device__ __forceinline__ void stage_left(const float* __restrict__ Fr,
                                           const float* __restrict__ Fi,
                                           float fsgn, const float* sR,
                                           const float* sI, float* dR,
                                           float* dI, int twsgn) {
  const int wave = threadIdx.x >> 5;
  const int L = threadIdx.x & 31;
  const int nn = L & 15;
  for (int t = wave; t < 64; t += 8) {
    const int rb = (t >> 3) << 4;
    const int cb = (t & 7) << 4;
    v8f cr = {};
    v8f ci = {};
    for (int kk = 0; kk < RAD; kk += 4) {
      v2f ar = frag_A(Fr, rb, kk, RAD, 1.0f);
      v2f ai = frag_A(Fi, rb, kk, RAD, fsgn);
      v2f br = frag_B(sR, kk, cb, RAD, 1.0f);
      v2f bi = frag_B(sI, kk, cb, RAD, 1.0f);
      cr = wmma4(ar, br, cr);
      cr = wmma4(-ai, bi, cr);  // cr += ArBr - AiBi
      ci = wmma4(ar, bi, ci);
      ci = wmma4(ai, br, ci);   // ci += ArBi + AiBr
    }
    float* crf = (float*)&cr;
    float* cif = (float*)&ci;
#pragma unroll
    for (int j = 0; j < 8; ++j) {
      const int mg = rb + j + ((L >> 4) << 3);
      const int ng = cb + nn;
      float re = crf[j], im = cif[j];
      if (twsgn != 0) {
        float ang = (float)twsgn * TWOPI_N * (float)((mg * ng) & (NFFT - 1));
        float s, c;
        __sincosf(ang, &s, &c);
        float nr = re * c - im * s;
        im = re * s + im * c;
        re = nr;
      }
      dR[mg * RAD + ng] = re;
      dI[mg * RAD + ng] = im;
    }
  }
  __syncthreads();
}

// ---- Stage: C = X * F~ (right multiply), optional twiddle / K-hat mult -----
__device__ __forceinline__ void stage_right(const float* __restrict__ Fr,
                                            const float* __restrict__ Fi,
                                            float fsgn, const float* sR,
                                            const float* sI, float* dR,
                                            float* dI, int twsgn,
                                            const float2* __restrict__ kmul) {
  const int wave = threadIdx.x >> 5;
  const int L = threadIdx.x & 31;
  const int nn = L & 15;
  for (int t = wave; t < 64; t += 8) {
    const int rb = (t >> 3) << 4;
    const int cb = (t & 7) << 4;
    v8f cr = {};
    v8f ci = {};
    for (int kk = 0; kk < RAD; kk += 4) {
      v2f ar = frag_A(sR, rb, kk, RAD, 1.0f);
      v2f ai = frag_A(sI, rb, kk, RAD, 1.0f);
      v2f br = frag_B(Fr, kk, cb, RAD, 1.0f);
      v2f bi = frag_B(Fi, kk, cb, RAD, fsgn);
      cr = wmma4(ar, br, cr);
      cr = wmma4(-ai, bi, cr);
      ci = wmma4(ar, bi, ci);
      ci = wmma4(ai, br, ci);
    }
    float* crf = (float*)&cr;
    float* cif = (float*)&ci;
#pragma unroll
    for (int j = 0; j < 8; ++j) {
      const int mg = rb + j + ((L >> 4) << 3);
      const int ng = cb + nn;
      float re = crf[j], im = cif[j];
      if (twsgn != 0) {
        float ang = (float)twsgn * TWOPI_N * (float)((mg * ng) & (NFFT - 1));
        float s, c;
        __sincosf(ang, &s, &c);
        float nr = re * c - im * s;
        im = re * s + im * c;
        re = nr;
      }
      if (kmul) {  // Uhat <- Uhat * Khat  (Khat already contains 1/NFFT)
        float2 kv = kmul[mg * RAD + ng];
        float nr = re * kv.x - im * kv.y;
        im = re * kv.y + im * kv.x;
        re = nr;
      }
      dR[mg * RAD + ng] = re;
      dI[mg * RAD + ng] = im;
    }
  }
  __syncthreads();
}

// ---- Forward stage 2 for the filter: C = X*F, scale, store Khat -----------
__device__ __forceinline__ void stage_store_khat(const float* __restrict__ Fr,
                                                 const float* __restrict__ Fi,
                                                 const float* sR,
                                                 const float* sI,
                                                 float2* __restrict__ khat) {
  const int wave = threadIdx.x >> 5;
  const int L = threadIdx.x & 31;
  const int nn = L & 15;
  const float sc = 1.0f / (float)NFFT;
  for (int t = wave; t < 64; t += 8) {
    const int rb = (t >> 3) << 4;
    const int cb = (t & 7) << 4;
    v8f cr = {};
    v8f ci = {};
    for (int kk = 0; kk < RAD; kk += 4) {
      v2f ar = frag_A(sR, rb, kk, RAD, 1.0f);
      v2f ai = frag_A(sI, rb, kk, RAD, 1.0f);
      v2f br = frag_B(Fr, kk, cb, RAD, 1.0f);
      v2f bi = frag_B(Fi, kk, cb, RAD, 1.0f);
      cr = wmma4(ar, br, cr);
      cr = wmma4(-ai, bi, cr);
      ci = wmma4(ar, bi, ci);
      ci = wmma4(ai, br, ci);
    }
    float* crf = (float*)&cr;
    float* cif = (float*)&ci;
#pragma unroll
    for (int j = 0; j < 8; ++j) {
      const int mg = rb + j + ((L >> 4) << 3);
      const int ng = cb + nn;
      float2 o;
      o.x = crf[j] * sc;
      o.y = cif[j] * sc;
      khat[mg * RAD + ng] = o;
    }
  }
}

// ---- Inverse stage 2 + epilogue: real part only, rows n<8192, y+bias*u ----
__device__ __forceinline__ void stage_final(const float* __restrict__ Fr,
                                            const float* __restrict__ Fi,
                                            const float* sR, const float* sI,
                                            const float* __restrict__ u,
                                            float dd, float* __restrict__ out) {
  const int wave = threadIdx.x >> 5;
  const int L = threadIdx.x & 31;
  const int nn = L & 15;
  for (int t = wave; t < 32; t += 8) {  // only b < 64 -> n < 8192
    const int rb = (t >> 3) << 4;
    const int cb = (t & 7) << 4;
    v8f cr = {};
    for (int kk = 0; kk < RAD; kk += 4) {
      v2f ar = frag_A(Fr, rb, kk, RAD, 1.0f);
      v2f ai = frag_A(Fi, rb, kk, RAD, -1.0f);  // conj(F)
      v2f br = frag_B(sR, kk, cb, RAD, 1.0f);
      v2f bi = frag_B(sI, kk, cb, RAD, 1.0f);
      cr = wmma4(ar, br, cr);
      cr = wmma4(-ai, bi, cr);  // Re only: output is real
    }
    float* crf = (float*)&cr;
#pragma unroll
    for (int j = 0; j < 8; ++j) {
      const int mg = rb + j + ((L >> 4) << 3);  // b
      const int ng = cb + nn;                   // a
      const int n = mg * RAD + ng;              // sample index
      out[n] = crf[j] + dd * u[n];
    }
  }
}

// ---------------------------------------------------------------------------
// K0: DFT matrix tables  F[i][j] = exp(-2*pi*i*j/128)  (exact arg reduction)
// ---------------------------------------------------------------------------
__global__ void __launch_bounds__(256) tables_kernel(float* __restrict__ Fr,
                                                     float* __restrict__ Fi) {
  int idx = blockIdx.x * 256 + threadIdx.x;
  if (idx >= RAD * RAD) return;
  int i = idx >> 7, j = idx & (RAD - 1);
  float ang = TWOPI_R * (float)((i * j) & (RAD - 1));
  Fr[idx] = __cosf(ang);
  Fi[idx] = -__sinf(ang);
}

// ---------------------------------------------------------------------------
// K1: implicit filter  kfilt[768][8192]
// block = 256 threads = 16 positions x 16 feature-lanes; MLP in LDS,
// final 64->768 projection as f32 WMMA, fused exponential modulation.
// ---------------------------------------------------------------------------
__global__ void __launch_bounds__(256) filter_kernel(
    const float* __restrict__ W0, const float* __restrict__ b0,
    const float* __restrict__ W1, const float* __restrict__ b1,
    const float* __restrict__ W2, const float* __restrict__ b2,
    const float* __restrict__ Wf, const float* __restrict__ freq,
    float* __restrict__ kfilt) {
  __shared__ float hA[16 * FO];
  __shared__ float hB[16 * FO];
  const int tid = threadIdx.x;
  const int p = tid >> 4;
  const int f0 = tid & 15;
  const int p0 = blockIdx.x << 4;
  const int pg = p0 + p;

  // positional embedding z[33] = [t, Re(zc), Im(zc)]
  float z[EMBD];
  z[0] = (float)pg / (float)(LSEQ - 1);
  const float w = TWOPI * (float)pg / (float)LSEQ;
#pragma unroll
  for (int b = 0; b < BANDS; ++b) {
    float fb = 1e-4f + (float)b * (((float)(BANDS - 1)) - 1e-4f) /
                           (float)(BANDS - 1);
    float s, c;
    __sincosf(fb * w, &s, &c);
    z[1 + b] = c;           // Re exp(-i f w)
    z[1 + BANDS + b] = -s;  // Im exp(-i f w)
  }
  // layer 0: 33 -> 64
#pragma unroll
  for (int q = 0; q < 4; ++q) {
    int ff = f0 + (q << 4);
    float acc = b0[ff];
    for (int e = 0; e < EMBD; ++e) acc += z[e] * W0[e * FO + ff];
    hA[p * FO + ff] = __sinf(freq[ff] * acc);
  }
  __syncthreads();
  // layer 1: 64 -> 64
#pragma unroll
  for (int q = 0; q < 4; ++q) {
    int ff = f0 + (q << 4);
    float acc = b1[ff];
    for (int e = 0; e < FO; ++e) acc += hA[p * FO + e] * W1[e * FO + ff];
    hB[p * FO + ff] = __sinf(freq[ff] * acc);
  }
  __syncthreads();
  // layer 2: 64 -> 64
#pragma unroll
  for (int q = 0; q < 4; ++q) {
    int ff = f0 + (q << 4);
    float acc = b2[ff];
    for (int e = 0; e < FO; ++e) acc += hB[p * FO + e] * W2[e * FO + ff];
    hA[p * FO + ff] = __sinf(freq[ff] * acc);
  }
  __syncthreads();

  // projection: [16 x 64] @ [64 x 768] with WMMA, + modulation
  const int wave = tid >> 5;
  const int L = tid & 31;
  const int nn = L & 15;
  for (int ct = wave; ct < DIMC / 16; ct += 8) {
    const int cb = ct << 4;
    v8f cr = {};
    for (int kk = 0; kk < FO; kk += 4) {
      v2f a = frag_A(hA, 0, kk, FO, 1.0f);
      v2f b = frag_B(Wf, kk, cb, DIMC, 1.0f);
      cr = wmma4(a, b, cr);
    }
    float* crf = (float*)&cr;
#pragma unroll
    for (int j = 0; j < 8; ++j) {
      const int m = j + ((L >> 4) << 3);   // local position
      const int d = cb + nn;               // channel
      const float tt = (float)(p0 + m) / (float)(LSEQ - 1);
      const float delta =
          MIN_AD + (float)d * (MAX_AD - MIN_AD) / (float)(DIMC - 1);
      kfilt[d * LSEQ + p0 + m] = crf[j] * (__expf(-tt * delta) + 0.05f);
    }
  }
}

// ---------------------------------------------------------------------------
// K2: forward FFT of each filter channel -> Khat (scaled by 1/NFFT)
// ---------------------------------------------------------------------------
__global__ void __launch_bounds__(256) fft_fwd_k_kernel(
    const float* __restrict__ Fr, const float* __restrict__ Fi,
    const float* __restrict__ kfilt, float2* __restrict__ Khat) {
  extern __shared__ float smem[];
  float* Xr = smem;
  float* Xi = smem + NFFT;
  float* Yr = smem + 2 * NFFT;
  float* Yi = smem + 3 * NFFT;
  const float* sig = kfilt + (size_t)blockIdx.x * LSEQ;

  // async DMA of the signal into LDS, zero-pad + imag plane via DS stores
  async_copy_to_lds(Xr, sig, LSEQ);
  for (int idx = threadIdx.x; idx < NFFT; idx += 256) {
    if (idx >= LSEQ) Xr[idx] = 0.0f;
    Xi[idx] = 0.0f;
  }
  wait_async();
  __syncthreads();

  stage_left(Fr, Fi, 1.0f, Xr, Xi, Yr, Yi, /*tw*/ -1);
  stage_store_khat(Fr, Fi, Yr, Yi, Khat + (size_t)blockIdx.x * NFFT);
}

// ---------------------------------------------------------------------------
// K3: fused forward-FFT(u) * Khat -> inverse-FFT -> y + bias*u
// ---------------------------------------------------------------------------
__global__ void __launch_bounds__(256) fftconv_kernel(
    const float* __restrict__ Fr, const float* __restrict__ Fi,
    const float* __restrict__ x, const float2* __restrict__ Khat,
    const float* __restrict__ bias, float* __restrict__ out) {
  extern __shared__ float smem[];
  float* Xr = smem;
  float* Xi = smem + NFFT;
  float* Yr = smem + 2 * NFFT;
  float* Yi = smem + 3 * NFFT;
  const int id = blockIdx.x;          // id = b*DIMC + d
  const int d = id % DIMC;
  const float* u = x + (size_t)id * LSEQ;
  const float2* kh = Khat + (size_t)d * NFFT;

  // warm Khat into cache while we FFT (global_prefetch_b8)
  {
    const char* kp = (const char*)kh;
    for (int off = (int)threadIdx.x * 128; off < NFFT * 8; off += 256 * 128)
      __builtin_prefetch(kp + off, 0, 1);
  }
  // async DMA of u into LDS, zero-pad + imag plane via DS stores
  async_copy_to_lds(Xr, u, LSEQ);
  for (int idx = threadIdx.x; idx < NFFT; idx += 256) {
    if (idx >= LSEQ) Xr[idx] = 0.0f;
    Xi[idx] = 0.0f;
  }
  wait_async();
  __syncthreads();

  // forward: Y = tw . (F * X);  X = (Y * F) . Khat
  stage_left(Fr, Fi, 1.0f, Xr, Xi, Yr, Yi, /*tw*/ -1);
  stage_right(Fr, Fi, 1.0f, Yr, Yi, Xr, Xi, /*tw*/ 0, kh);
  // inverse: Y = tw* . (X * conj(F));  out = Re(conj(F) * Y) + bias*u
  stage_right(Fr, Fi, -1.0f, Xr, Xi, Yr, Yi, /*tw*/ +1, nullptr);
  stage_final(Fr, Fi, Yr, Yi, u, bias[d], out + (size_t)id * LSEQ);
}

// ---------------------------------------------------------------------------
// Host launcher
// ---------------------------------------------------------------------------
extern "C" void kernel_launch(void* const* d_in, const int* in_sizes, int n_in,
                              void* d_out, int out_size, void* d_ws,
                              size_t ws_size, hipStream_t stream) {
  (void)in_sizes; (void)n_in; (void)out_size; (void)ws_size;
  const float* x    = (const float*)d_in[0];
  const float* W0   = (const float*)d_in[1];
  const float* b0   = (const float*)d_in[2];
  const float* W1   = (const float*)d_in[3];
  const float* b1   = (const float*)d_in[4];
  const float* W2   = (const float*)d_in[5];
  const float* b2   = (const float*)d_in[6];
  const float* Wf   = (const float*)d_in[7];
  const float* freq = (const float*)d_in[8];
  const float* bias = (const float*)d_in[9];
  float* out = (float*)d_out;

  // workspace layout (floats): Fr | Fi | kfilt | Khat   (~120 MB total)
  float* ws = (float*)d_ws;
  float* Fr = ws;
  float* Fi = ws + RAD * RAD;
  float* kfilt = ws + 2 * RAD * RAD;
  float2* Khat = (float2*)(ws + 2 * RAD * RAD + (size_t)DIMC * LSEQ);

  const int lds_bytes = 4 * NFFT * (int)sizeof(float);  // 262144
  hipFuncSetAttribute((const void*)fft_fwd_k_kernel,
                      hipFuncAttributeMaxDynamicSharedMemorySize, lds_bytes);
  hipFuncSetAttribute((const void*)fftconv_kernel,
                      hipFuncAttributeMaxDynamicSharedMemorySize, lds_bytes);

  tables_kernel<<<(RAD * RAD + 255) / 256, 256, 0, stream>>>(Fr, Fi);
  filter_kernel<<<LSEQ / 16, 256, 0, stream>>>(W0, b0, W1, b1, W2, b2, Wf,
                                               freq, kfilt);
  fft_fwd_k_kernel<<<DIMC, 256, lds_bytes, stream>>>(Fr, Fi, kfilt, Khat);
  fftconv_kernel<<<2 * DIMC, 256, lds_bytes, stream>>>(Fr, Fi, x, Khat, bias,
                                                       out);
}